// QuaternionAnchorLayer_68702296867232
// MI455X (gfx1250) — compile-verified
//
#include <hip/hip_runtime.h>

// ---------------------------------------------------------------------------
// QuaternionAnchorLayer for MI455X (gfx1250, wave32).
//
// Shapes: B=2, N=2048, D=256 (=4*Q, Q=64), K=64 anchors, H=4 heads, hd=64.
//
// Math notes:
//  * d[b,k,n] = sum_D hamilton(p, conj(a))^2.  Quaternion norm is
//    multiplicative (|p*q| = |p||q|), so this is exactly
//    sum_q |p_q|^2 * |a_q|^2 -- a [BN,Q]x[Q,K] dot instead of a [B,K,N,D]
//    hamilton tensor.
//  * qlinear(x,w,b) = x @ W^T + b with W the 256x256 Hamilton block matrix.
//    We materialize Wt[i][o] = W[o][i] once per weight set, then all five
//    qlinears are fp32 GEMMs on V_WMMA_F32_16X16X4_F32 with the X tile
//    staged through LDS and 2-way A-fragment reuse.
// ---------------------------------------------------------------------------

typedef __attribute__((ext_vector_type(2))) float v2f;
typedef __attribute__((ext_vector_type(8))) float v8f;

#define B_  2
#define N_  2048
#define D_  256
#define K_  64
#define QQ  64      // D/4
#define H_  4
#define HD  64

// ---------------------------------------------------------------------------
// Build Wt (256x256, row = input dim i, col = output dim o) from w[4,64,64].
//   W = [[ r,-i,-j,-k],[ i, r,-k, j],[ j, k, r,-i],[ k,-j, i, r]]
// ---------------------------------------------------------------------------
__global__ void build_wt_kernel(const float* __restrict__ w,
                                float* __restrict__ Wt) {
  int e = blockIdx.x * 256 + threadIdx.x;       // 0 .. 65535
  int i = e >> 8;                               // input dim
  int o = e & 255;                              // output dim
  int bo = o >> 6, bi = i >> 6, p = o & 63, q = i & 63;
  const int   comp[4][4] = {{0,1,2,3},{1,0,3,2},{2,3,0,1},{3,2,1,0}};
  const float sgn [4][4] = {{ 1.f,-1.f,-1.f,-1.f},
                            { 1.f, 1.f,-1.f, 1.f},
                            { 1.f, 1.f, 1.f,-1.f},
                            { 1.f,-1.f, 1.f, 1.f}};
  Wt[e] = sgn[bo][bi] * w[comp[bo][bi] * 4096 + p * 64 + q];
}

// ---------------------------------------------------------------------------
// Per-quaternion squared norms of q_rot rows (-> P[BN,64]) and anchors
// rows (-> A[K,64]).  Row layout of a D=256 vector: [r(64) i(64) j(64) k(64)].
// ---------------------------------------------------------------------------
__global__ void sqnorm_kernel(const float* __restrict__ q_rot,
                              const float* __restrict__ anchors,
                              float* __restrict__ P,
                              float* __restrict__ A) {
  int idx = blockIdx.x * 256 + threadIdx.x;     // (BN + K) * 64 elements
  int row = idx >> 6, q = idx & 63;
  const float* src;
  float* dst;
  if (row < B_ * N_) {
    src = q_rot + (size_t)row * D_;
    dst = P + (size_t)row * QQ;
  } else {
    int r = row - B_ * N_;
    if (r >= K_) return;
    src = anchors + (size_t)r * D_;
    dst = A + (size_t)r * QQ;
  }
  float a = src[q], b = src[q + 64], c = src[q + 128], d = src[q + 192];
  dst[q] = a * a + b * b + c * c + d * d;
}

// ---------------------------------------------------------------------------
// incid[b,k,n] = exp(-d/sigma_sq[k]) * mask[b,n],  denom[b,k] = max(sum,1e-6).
// One block per (b,k); 256 threads each cover 8 values of n.
// ---------------------------------------------------------------------------
__global__ void __launch_bounds__(256)
incid_kernel(const float* __restrict__ P, const float* __restrict__ A,
             const float* __restrict__ log_sigma,
             const float* __restrict__ mask,
             float* __restrict__ incid, float* __restrict__ denom) {
  int b = blockIdx.x >> 6, k = blockIdx.x & 63;
  __shared__ float Arow[QQ];
  __shared__ float red[256];
  int t = threadIdx.x;
  if (t < QQ) Arow[t] = A[k * QQ + t];
  __syncthreads();

  float s  = expf(log_sigma[k]);
  float ss = fmaxf(s * s, 1e-6f);
  float inv = 1.0f / ss;

  float psum = 0.f;
  for (int j = 0; j < N_ / 256; ++j) {
    int n = j * 256 + t;
    const float* pr = P + (size_t)(b * N_ + n) * QQ;
    float d = 0.f;
#pragma unroll 16
    for (int q = 0; q < QQ; ++q) d += pr[q] * Arow[q];
    float v = expf(-d * inv) * mask[b * N_ + n];
    incid[(size_t)(b * K_ + k) * N_ + n] = v;
    psum += v;
  }
  red[t] = psum;
  __syncthreads();
  for (int sredux = 128; sredux > 0; sredux >>= 1) {
    if (t < sredux) red[t] += red[t + sredux];
    __syncthreads();
  }
  if (t == 0) denom[b * K_ + k] = fmaxf(red[0], 1e-6f);
}

// ---------------------------------------------------------------------------
// hraw[b,k,d] = (sum_n incid[b,k,n] * q_rot[b,n,d]) / denom[b,k].
// One block per (b,k); thread = output column d; incid staged through LDS.
// ---------------------------------------------------------------------------
__global__ void __launch_bounds__(256)
hmean_kernel(const float* __restrict__ incid, const float* __restrict__ denom,
             const float* __restrict__ q_rot, float* __restrict__ hraw) {
  int b = blockIdx.x >> 6, k = blockIdx.x & 63;
  int d = threadIdx.x;
  __shared__ float inc_s[256];
  float acc = 0.f;
  const float* ibase = incid + (size_t)(b * K_ + k) * N_;
  for (int c = 0; c < N_ / 256; ++c) {
    __syncthreads();
    inc_s[d] = ibase[c * 256 + d];
    __syncthreads();
    const float* xb = q_rot + ((size_t)b * N_ + c * 256) * D_ + d;
#pragma unroll 8
    for (int j = 0; j < 256; ++j) acc += inc_s[j] * xb[(size_t)j * D_];
  }
  hraw[(size_t)(b * K_ + k) * D_ + d] = acc / denom[b * K_ + k];
}

// ---------------------------------------------------------------------------
// Y[M,256] = X[M,256] @ Wt + bias  on V_WMMA_F32_16X16X4_F32.
//
// Block = 256 threads = 8 waves; blockIdx.x owns one 16-row M tile.
// The X tile (16x256) is staged into LDS once (coalesced, row pitch 260
// floats -> conflict-free b64 A-fragment reads: low half-wave hits banks
// 4l/4l+1, high half 4l+2/4l+3).  Each wave computes TWO adjacent 16-col
// N tiles, reusing each A fragment for 2 WMMAs per k-step.
//
// ISA 7.12.2 fp32 layouts (wave32):
//  A 16x4 : lane l -> m = l&15;  VGPR0 = K(2*(l>>4)),   VGPR1 = K(2*(l>>4)+1)
//  B 4x16 : lane l -> n = l&15;  VGPR0 = row 2*(l>>4),  VGPR1 = row 2*(l>>4)+1
//  C/D    : VGPR v -> row v (lanes 0-15) / row v+8 (lanes 16-31), col = l&15
// ---------------------------------------------------------------------------
#define XPITCH 260
__global__ void __launch_bounds__(256)
qlinear_wmma_kernel(const float* __restrict__ X, const float* __restrict__ Wt,
                    const float* __restrict__ bias, float* __restrict__ Y) {
  __shared__ float Xs[16 * XPITCH];
  int tid = threadIdx.x;
  int m0 = blockIdx.x * 16;

  // Cooperative coalesced stage of the 16x256 X tile.
  {
    const float* xb = X + (size_t)m0 * 256 + tid;
#pragma unroll
    for (int r = 0; r < 16; ++r) Xs[r * XPITCH + tid] = xb[(size_t)r * 256];
  }
  __syncthreads();

  int wave = tid >> 5;                // 0..7 -> cols [wave*32, wave*32+32)
  int lane = tid & 31;
  int n0 = wave * 32;
  int mi = lane & 15;                 // m for A, n for B/C/D
  int kh = lane >> 4;

  const float* xrow = Xs + mi * XPITCH;
  v8f acc0 = {}, acc1 = {};
  for (int kk = 0; kk < 256; kk += 4) {
    const float* wr0 = Wt + (size_t)(kk + 2 * kh) * 256;
    // Warm L2/L0 for the B panel 8 k-steps ahead (global_prefetch_b8).
    if (kk + 32 < 256) __builtin_prefetch(wr0 + 32 * 256 + n0 + mi, 0, 1);
    v2f a, b0, b1;
    a.x  = xrow[kk + 2 * kh];
    a.y  = xrow[kk + 2 * kh + 1];
    b0.x = wr0[n0 + mi];
    b0.y = wr0[256 + n0 + mi];
    b1.x = wr0[n0 + 16 + mi];
    b1.y = wr0[256 + n0 + 16 + mi];
    acc0 = __builtin_amdgcn_wmma_f32_16x16x4_f32(
        false, a, false, b0, (short)0, acc0, false, false);
    acc1 = __builtin_amdgcn_wmma_f32_16x16x4_f32(
        false, a, false, b1, (short)0, acc1, false, false);
  }
  int col0 = n0 + mi, col1 = n0 + 16 + mi;
  float bv0 = bias[col0], bv1 = bias[col1];
#pragma unroll
  for (int v = 0; v < 8; ++v) {
    int row = m0 + v + 8 * kh;
    Y[(size_t)row * 256 + col0] = acc0[v] + bv0;
    Y[(size_t)row * 256 + col1] = acc1[v] + bv1;
  }
}

// ---------------------------------------------------------------------------
// Attention: scores = Qm.Km^T / 8, softmax over K=64, out = attn @ Vm.
// Block = 256 threads = 4 subgroups of 64; each block owns (b,h) and 32 n's;
// Km/Vm head tiles (64x64) staged in LDS once per block.
// ---------------------------------------------------------------------------
__global__ void __launch_bounds__(256)
attn_kernel(const float* __restrict__ Qm, const float* __restrict__ Km,
            const float* __restrict__ Vm, float* __restrict__ out) {
  const int nb_per = N_ / 32;                       // 64 blocks per (b,h)
  int bid = blockIdx.x;
  int b   = bid / (H_ * nb_per);
  int rem = bid % (H_ * nb_per);
  int h   = rem / nb_per;
  int nb  = rem % nb_per;

  __shared__ float Km_s[64][65];
  __shared__ float Vm_s[64][65];
  __shared__ float sc[4][64];
  __shared__ float sp[4][64];

  int tid = threadIdx.x;
  for (int e = tid; e < 64 * 64; e += 256) {
    int k = e >> 6, dd = e & 63;
    Km_s[k][dd] = Km[(size_t)(b * K_ + k) * D_ + h * HD + dd];
    Vm_s[k][dd] = Vm[(size_t)(b * K_ + k) * D_ + h * HD + dd];
  }
  __syncthreads();

  int sg = tid >> 6;                                // subgroup 0..3
  int t  = tid & 63;                                // k index, then d index
  const float scale = 0.125f;                       // 1/sqrt(64)

  for (int it = 0; it < 8; ++it) {
    int n = nb * 32 + it * 4 + sg;
    const float* qr = Qm + (size_t)(b * N_ + n) * D_ + h * HD;
    float s = 0.f;
#pragma unroll 16
    for (int dd = 0; dd < 64; ++dd) s += qr[dd] * Km_s[t][dd];
    s *= scale;
    sc[sg][t] = s;
    __syncthreads();
    float mx = -3.402823466e38f;
    for (int j = 0; j < 64; ++j) mx = fmaxf(mx, sc[sg][j]);
    float p = expf(s - mx);
    sp[sg][t] = p;
    __syncthreads();
    float sum = 0.f;
    for (int j = 0; j < 64; ++j) sum += sp[sg][j];
    float o = 0.f;
#pragma unroll 16
    for (int k = 0; k < 64; ++k) o += sp[sg][k] * Vm_s[k][t];
    out[(size_t)(b * N_ + n) * D_ + h * HD + t] = o / sum;
    __syncthreads();
  }
}

// ---------------------------------------------------------------------------
// Host-side orchestration.
// ---------------------------------------------------------------------------
extern "C" void kernel_launch(void* const* d_in, const int* in_sizes, int n_in,
                              void* d_out, int out_size, void* d_ws,
                              size_t ws_size, hipStream_t stream) {
  const float* q_rot     = (const float*)d_in[0];
  const float* mask      = (const float*)d_in[1];
  const float* anchors   = (const float*)d_in[2];
  const float* log_sigma = (const float*)d_in[3];
  const float* aggr_w    = (const float*)d_in[4];
  const float* aggr_b    = (const float*)d_in[5];
  const float* fcq_w     = (const float*)d_in[6];
  const float* fcq_b     = (const float*)d_in[7];
  const float* fck_w     = (const float*)d_in[8];
  const float* fck_b     = (const float*)d_in[9];
  const float* fcv_w     = (const float*)d_in[10];
  const float* fcv_b     = (const float*)d_in[11];
  const float* fco_w     = (const float*)d_in[12];
  const float* fco_b     = (const float*)d_in[13];

  // Workspace layout (floats), ~12.3 MB total.
  float* ws    = (float*)d_ws;
  float* Wt    = ws;                        // 5 * 65536
  float* P     = Wt    + 5 * 65536;         // BN * 64   = 262144
  float* An    = P     + 262144;            // K * 64    = 4096
  float* incid = An    + 4096;              // B*K*N     = 262144
  float* denom = incid + 262144;            // B*K       = 128
  float* hraw  = denom + 128;               // B*K*D     = 32768
  float* h     = hraw  + 32768;             // 32768
  float* Qm    = h     + 32768;             // BN*D      = 1048576
  float* Km    = Qm    + 1048576;           // 32768
  float* Vm    = Km    + 32768;             // 32768
  float* aout  = Vm    + 32768;             // 1048576
  (void)ws_size; (void)n_in; (void)in_sizes; (void)out_size;

  // 1. Hamilton block matrices (transposed) for all five weight sets.
  const float* wsrc[5] = {aggr_w, fcq_w, fck_w, fcv_w, fco_w};
  for (int wi = 0; wi < 5; ++wi)
    build_wt_kernel<<<256, 256, 0, stream>>>(wsrc[wi], Wt + wi * 65536);

  // 2. Per-quaternion squared norms (norm-multiplicativity shortcut).
  sqnorm_kernel<<<(B_ * N_ + K_) * 64 / 256, 256, 0, stream>>>(q_rot, anchors,
                                                               P, An);

  // 3. Gaussian incidence + denominators.
  incid_kernel<<<B_ * K_, 256, 0, stream>>>(P, An, log_sigma, mask, incid,
                                            denom);

  // 4. Masked weighted mean -> hraw.
  hmean_kernel<<<B_ * K_, 256, 0, stream>>>(incid, denom, q_rot, hraw);

  // 5. Quaternion linears on the WMMA pipe.
  qlinear_wmma_kernel<<<(B_ * K_) / 16, 256, 0, stream>>>(hraw, Wt + 0 * 65536,
                                                          aggr_b, h);
  qlinear_wmma_kernel<<<(B_ * N_) / 16, 256, 0, stream>>>(q_rot, Wt + 1 * 65536,
                                                          fcq_b, Qm);
  qlinear_wmma_kernel<<<(B_ * K_) / 16, 256, 0, stream>>>(h, Wt + 2 * 65536,
                                                          fck_b, Km);
  qlinear_wmma_kernel<<<(B_ * K_) / 16, 256, 0, stream>>>(h, Wt + 3 * 65536,
                                                          fcv_b, Vm);

  // 6. Quaternion MHA.
  attn_kernel<<<B_ * H_ * (N_ / 32), 256, 0, stream>>>(Qm, Km, Vm, aout);

  // 7. Output projection straight into d_out.
  qlinear_wmma_kernel<<<(B_ * N_) / 16, 256, 0, stream>>>(aout, Wt + 4 * 65536,
                                                          fco_b,
                                                          (float*)d_out);
}